// MultiHeadGraphAttention_15058155340491
// MI455X (gfx1250) — compile-verified
//
#include <hip/hip_runtime.h>
#include <hip/hip_bf16.h>
#include <math.h>

typedef __attribute__((ext_vector_type(2))) float v2f;
typedef __attribute__((ext_vector_type(8))) float v8f;

#define N_NODES 50000
#define N_EDGES 800000
#define IN_DIM 128
#define UNITS 32
#define HEADS 8
#define OUT_DIM 256           // HEADS * UNITS
#define LEAKY_ALPHA 0.2f

// ---------------------------------------------------------------------------
// float atomic max via sign-split integer atomics (works with -inf init)
// ---------------------------------------------------------------------------
__device__ __forceinline__ void atomicMaxF(float* addr, float val) {
  if (val >= 0.0f) atomicMax((int*)addr, __float_as_int(val));
  else             atomicMin((unsigned int*)addr, __float_as_uint(val));
}

// ---------------------------------------------------------------------------
// 0) init: smax = -inf
// ---------------------------------------------------------------------------
__global__ __launch_bounds__(256) void gat_init_neginf(float* __restrict__ smax, int n) {
  int i = blockIdx.x * 256 + threadIdx.x;
  if (i < n) smax[i] = -INFINITY;
}

// ---------------------------------------------------------------------------
// 1) XP = X @ W using V_WMMA_F32_16X16X4_F32.  One wave per 16x16 C tile.
//    A (16x4 f32): lane 0-15 -> M=lane, VGPR0=K0,VGPR1=K1; lane 16-31 -> K2,K3
//    B (4x16 f32): lane 0-15 -> N=lane, VGPR0=K0,VGPR1=K1; lane 16-31 -> K2,K3
//    C/D: VGPR i: lanes 0-15 -> M=i, lanes 16-31 -> M=i+8; N = lane&15
// ---------------------------------------------------------------------------
__global__ __launch_bounds__(256) void gat_gemm_wmma(const float* __restrict__ X,
                                                     const float* __restrict__ W,
                                                     float* __restrict__ XP) {
  const int gwave = (blockIdx.x * 256 + (int)threadIdx.x) >> 5;
  const int lane  = threadIdx.x & 31;
  const int tilesN = OUT_DIM / 16;                 // 16
  const int tileM  = gwave / tilesN;               // 0..3124
  const int tileN  = gwave % tilesN;               // 0..15
  if (tileM >= N_NODES / 16) return;               // exact grid: never taken

  const int half = lane >> 4;                      // 0 or 1 (selects K pair)
  const int l16  = lane & 15;
  const int row  = tileM * 16 + l16;               // M index for A
  const int col  = tileN * 16 + l16;               // N index for B

  const float* ap = X + (size_t)row * IN_DIM + half * 2;   // contiguous K pair
  const float* bp = W + (size_t)(half * 2) * OUT_DIM + col;

  v8f acc = {};
#pragma unroll
  for (int k = 0; k < IN_DIM; k += 4) {
    v2f a = *(const v2f*)(ap + k);                 // X[row][k+2h .. k+2h+1]
    v2f b;
    b.x = bp[(size_t)k * OUT_DIM];                 // W[k+2h  ][col]
    b.y = bp[(size_t)(k + 1) * OUT_DIM];           // W[k+2h+1][col]
    acc = __builtin_amdgcn_wmma_f32_16x16x4_f32(
        /*neg_a=*/false, a, /*neg_b=*/false, b,
        /*c_mod=*/(short)0, acc, /*reuse_a=*/false, /*reuse_b=*/false);
  }

  float* op = XP + (size_t)(tileM * 16 + half * 8) * OUT_DIM + tileN * 16 + l16;
#pragma unroll
  for (int i = 0; i < 8; ++i) op[(size_t)i * OUT_DIM] = acc[i];
}

// ---------------------------------------------------------------------------
// 2) f_t[n,h] = <xp[n,h,:], att1[h,:]> ;  f_s likewise with att2
// ---------------------------------------------------------------------------
__global__ __launch_bounds__(256) void gat_attn_proj(const float* __restrict__ XP,
                                                     const float* __restrict__ att1,
                                                     const float* __restrict__ att2,
                                                     float* __restrict__ ft,
                                                     float* __restrict__ fs) {
  int idx = blockIdx.x * 256 + threadIdx.x;        // node*HEADS + h
  if (idx >= N_NODES * HEADS) return;
  int h = idx & (HEADS - 1);
  const float* v  = XP + (size_t)(idx >> 3) * OUT_DIM + h * UNITS;
  const float* a1 = att1 + h * UNITS;
  const float* a2 = att2 + h * UNITS;
  float s1 = 0.0f, s2 = 0.0f;
#pragma unroll
  for (int u = 0; u < UNITS; ++u) {
    float x = v[u];
    s1 += x * a1[u];
    s2 += x * a2[u];
  }
  ft[idx] = s1;
  fs[idx] = s2;
}

// ---------------------------------------------------------------------------
// 3) per-target segment max of leaky-relu scores
// ---------------------------------------------------------------------------
__global__ __launch_bounds__(256) void gat_edge_max(const int* __restrict__ edges,
                                                    const float* __restrict__ ft,
                                                    const float* __restrict__ fs,
                                                    float* __restrict__ smax) {
  int e = blockIdx.x * 256 + threadIdx.x;
  if (e >= N_EDGES) return;
  int s = edges[2 * e + 0];
  int t = edges[2 * e + 1];
#pragma unroll
  for (int h = 0; h < HEADS; ++h) {
    float sc = ft[t * HEADS + h] + fs[s * HEADS + h];
    sc = sc > 0.0f ? sc : LEAKY_ALPHA * sc;
    atomicMaxF(&smax[t * HEADS + h], sc);
  }
}

// ---------------------------------------------------------------------------
// 4) per-target segment sum of exp(score - max)
// ---------------------------------------------------------------------------
__global__ __launch_bounds__(256) void gat_edge_sum(const int* __restrict__ edges,
                                                    const float* __restrict__ ft,
                                                    const float* __restrict__ fs,
                                                    const float* __restrict__ smax,
                                                    float* __restrict__ ssum) {
  int e = blockIdx.x * 256 + threadIdx.x;
  if (e >= N_EDGES) return;
  int s = edges[2 * e + 0];
  int t = edges[2 * e + 1];
#pragma unroll
  for (int h = 0; h < HEADS; ++h) {
    float sc = ft[t * HEADS + h] + fs[s * HEADS + h];
    sc = sc > 0.0f ? sc : LEAKY_ALPHA * sc;
    atomicAdd(&ssum[t * HEADS + h], __expf(sc - smax[t * HEADS + h]));
  }
}

// ---------------------------------------------------------------------------
// 5) aggregation: one wave per edge; lane = unit, loop over heads.
//    out[t] += softmax_weight(e,h) * xp[s]  via f32 L2 atomics.
// ---------------------------------------------------------------------------
__global__ __launch_bounds__(256) void gat_aggregate(const int* __restrict__ edges,
                                                     const float* __restrict__ ft,
                                                     const float* __restrict__ fs,
                                                     const float* __restrict__ smax,
                                                     const float* __restrict__ ssum,
                                                     const float* __restrict__ XP,
                                                     float* __restrict__ out) {
  int e    = (blockIdx.x * 256 + (int)threadIdx.x) >> 5;
  int lane = threadIdx.x & 31;
  if (e >= N_EDGES) return;
  int s = edges[2 * e + 0];
  int t = edges[2 * e + 1];
  const float* src = XP + (size_t)s * OUT_DIM;
  float*       dst = out + (size_t)t * OUT_DIM;
#pragma unroll
  for (int h = 0; h < HEADS; ++h) {
    float sc = ft[t * HEADS + h] + fs[s * HEADS + h];     // scalar broadcast loads
    sc = sc > 0.0f ? sc : LEAKY_ALPHA * sc;
    float w = __expf(sc - smax[t * HEADS + h]) / ssum[t * HEADS + h];
    atomicAdd(dst + h * UNITS + lane, w * src[h * UNITS + lane]);
  }
}

// ---------------------------------------------------------------------------
// 6) out = elu(out + bias), in place
// ---------------------------------------------------------------------------
__global__ __launch_bounds__(256) void gat_bias_elu(float* __restrict__ out,
                                                    const float* __restrict__ bias) {
  int i = blockIdx.x * 256 + threadIdx.x;
  if (i >= N_NODES * OUT_DIM) return;
  float v = out[i] + bias[i & (OUT_DIM - 1)];
  out[i] = v > 0.0f ? v : (__expf(v) - 1.0f);
}

// ---------------------------------------------------------------------------
extern "C" void kernel_launch(void* const* d_in, const int* in_sizes, int n_in,
                              void* d_out, int out_size, void* d_ws, size_t ws_size,
                              hipStream_t stream) {
  const float* x     = (const float*)d_in[0];
  const int*   edges = (const int*)d_in[1];
  const float* kernW = (const float*)d_in[2];
  const float* att1  = (const float*)d_in[3];
  const float* att2  = (const float*)d_in[4];
  const float* bias  = (const float*)d_in[5];
  float* out = (float*)d_out;

  // workspace layout (floats): XP | ft | fs | smax | ssum  (~57.6 MB)
  float* XP   = (float*)d_ws;
  float* ft   = XP + (size_t)N_NODES * OUT_DIM;
  float* fs   = ft + (size_t)N_NODES * HEADS;
  float* smax = fs + (size_t)N_NODES * HEADS;
  float* ssum = smax + (size_t)N_NODES * HEADS;

  // 0) init accumulators (graph-capture safe: async memset + fill kernel)
  hipMemsetAsync(d_out, 0, (size_t)N_NODES * OUT_DIM * sizeof(float), stream);
  hipMemsetAsync(ssum, 0, (size_t)N_NODES * HEADS * sizeof(float), stream);
  gat_init_neginf<<<(N_NODES * HEADS + 255) / 256, 256, 0, stream>>>(smax, N_NODES * HEADS);

  // 1) projection GEMM (WMMA f32): 3125 x 16 tiles -> 50000 waves
  {
    int waves  = (N_NODES / 16) * (OUT_DIM / 16);
    int blocks = waves * 32 / 256;                 // 6250, exact
    gat_gemm_wmma<<<blocks, 256, 0, stream>>>(x, kernW, XP);
  }

  // 2) attention logits
  gat_attn_proj<<<(N_NODES * HEADS + 255) / 256, 256, 0, stream>>>(XP, att1, att2, ft, fs);

  // 3) segment max
  gat_edge_max<<<(N_EDGES + 255) / 256, 256, 0, stream>>>(edges, ft, fs, smax);

  // 4) segment sum of exp
  gat_edge_sum<<<(N_EDGES + 255) / 256, 256, 0, stream>>>(edges, ft, fs, smax, ssum);

  // 5) weighted scatter-add aggregation: one wave per edge
  gat_aggregate<<<(int)(((size_t)N_EDGES * 32) / 256), 256, 0, stream>>>(
      edges, ft, fs, smax, ssum, XP, out);

  // 6) bias + ELU
  gat_bias_elu<<<(N_NODES * OUT_DIM + 255) / 256, 256, 0, stream>>>(out, bias);
}